// QuantizedShaper_4157528343032
// MI455X (gfx1250) — compile-verified
//
#include <hip/hip_runtime.h>
#include <cstdint>
#include <cstddef>

// ---------------------------------------------------------------------------
// Problem constants (from reference): B=256, L=8192, DIM=256, NPAT=64,
// HIST=32, WIN=8  ->  T = (L + HIST-1 - HIST)/WIN + 1 = 1024
// ---------------------------------------------------------------------------
#define B_    256
#define L_    8192
#define DIM_  256
#define NPAT_ 64
#define HIST_ 32
#define WIN_  8
#define T_    1024

typedef __attribute__((ext_vector_type(2))) float    v2f;
typedef __attribute__((ext_vector_type(8))) float    v8f;
typedef __attribute__((ext_vector_type(4))) unsigned v4u;
typedef __attribute__((ext_vector_type(8))) int      v8i;
typedef __attribute__((ext_vector_type(4))) int      v4i;

// ---------------------------------------------------------------------------
// CDNA5 async helpers
// ---------------------------------------------------------------------------

// Per-lane async global->LDS copy of 16 bytes (GV addressing, tracked on
// ASYNCcnt). LDS destination address = low 32 bits of the generic pointer
// (generic shared pointers are {aperture_hi32, lds_offset32} on AMDGCN).
__device__ __forceinline__ void async_copy_b128(void* lds_dst, const void* gsrc) {
  unsigned l = (unsigned)(uintptr_t)lds_dst;
  asm volatile("global_load_async_to_lds_b128 %0, %1, off"
               :: "v"(l), "v"(gsrc)
               : "memory");
}

__device__ __forceinline__ void wait_async0() {
#if __has_builtin(__builtin_amdgcn_s_wait_asynccnt)
  __builtin_amdgcn_s_wait_asynccnt(0);
#else
  asm volatile("s_wait_asynccnt 0" ::: "memory");
#endif
}

// Tensor Data Mover: 1-D tile copy of 4096 f32 (16 KB) global -> LDS.
// D# group0: count=1, lds_addr, 57-bit global addr, type=2 ("image").
// D# group1: data_size=2 (4 B), tensor_dim0=4096, tensor_dim1=1,
//            tile_dim0=4096 (1-D: tile_dim1=0), tensor_dim0_stride=4096.
// Groups 2/3 unused (zero). Tracked on TENSORcnt.
__device__ __forceinline__ void tdm_load_chunk16k(void* lds_dst, const float* gsrc) {
  unsigned ldsa = (unsigned)(uintptr_t)lds_dst;
  unsigned long long ga = (unsigned long long)(uintptr_t)gsrc;
  v4u g0 = { 1u,                               // count=1, user mode
             ldsa,                             // lds_addr
             (unsigned)ga,                     // global_addr[31:0]
             ((unsigned)(ga >> 32) & 0x01FFFFFFu) | 0x80000000u }; // addr[56:32] | type=2
  v8i g1 = { 0x00020000,                       // data_size=2 (4 bytes)
             0x10000000,                       // tensor_dim0[15:0]=4096 at bits[63:48]
             0x00010000,                       // tensor_dim1[15:0]=1 at bits[95:80]
             0x10000000,                       // tile_dim0=4096 at bits[127:112]
             0,                                // tile_dim1=0, tile_dim2=0
             0x00001000,                       // tensor_dim0_stride[31:0]=4096
             0, 0 };
  v4i gz = { 0, 0, 0, 0 };
#if defined(__clang_major__) && (__clang_major__ >= 23)
  v8i gz8 = { 0, 0, 0, 0, 0, 0, 0, 0 };
  __builtin_amdgcn_tensor_load_to_lds(g0, g1, gz, gz, gz8, 0);
#else
  __builtin_amdgcn_tensor_load_to_lds(g0, g1, gz, gz, 0);
#endif
}

__device__ __forceinline__ void wait_tensor0() {
  __builtin_amdgcn_s_wait_tensorcnt((short)0);
}

// ---------------------------------------------------------------------------
// Kernel 1: fold conv_w/conv_b through keys_w.
//   ck[p][h] = sum_d keys_w[p,d] * conv_w[d,0,h]   (64 x 32)
//   cb[p]    = sum_d keys_w[p,d] * conv_b[d]
// ---------------------------------------------------------------------------
__global__ void __launch_bounds__(256)
prep_kernel(const float* __restrict__ keys_w,
            const float* __restrict__ conv_w,
            const float* __restrict__ conv_b,
            float* __restrict__ ck, float* __restrict__ cb) {
  const int tid = threadIdx.x;
  for (int i = tid; i < NPAT_ * HIST_; i += 256) {
    const int p = i >> 5, h = i & 31;
    float acc = 0.f;
    for (int d = 0; d < DIM_; ++d)
      acc += keys_w[p * DIM_ + d] * conv_w[d * HIST_ + h];
    ck[i] = acc;
  }
  for (int i = tid; i < NPAT_; i += 256) {
    float acc = 0.f;
    for (int d = 0; d < DIM_; ++d)
      acc += keys_w[i * DIM_ + d] * conv_b[d];
    cb[i] = acc;
  }
}

// ---------------------------------------------------------------------------
// Kernel 2: scores[b][t][p] = clip(cb[p] + sum_h xpad[b,8t+h]*ck[p][h], 0, 6)
// One workgroup per batch. x row staged in LDS via async b128 copies with a
// 32-float left pad (xpad[i] == lds_x2[i+1]; lds_x2[1..31] zeroed).
// GEMM via V_WMMA_F32_16X16X4_F32:  M=16 t-rows, N=16 patterns, K=4 (8 steps).
// Wave w handles t in [128w, 128w+128): 8 M-tiles x 4 N-tiles x 8 k-steps.
//
// f32 WMMA fragment layouts (ISA 7.12.2):
//   A 16x4 : lanes 0-15 -> M=lane, K={0(v0),1(v1)}; lanes 16-31 -> K={2,3}
//   B 4x16 : lanes 0-15 -> N=lane, K={0(v0),1(v1)}; lanes 16-31 -> K={2,3}
//   C/D    : VGPR i, lane l -> M = i + 8*(l/16), N = l%16
// ---------------------------------------------------------------------------
__global__ void __launch_bounds__(256)
scores_kernel(const float* __restrict__ x,
              const float* __restrict__ ck,
              const float* __restrict__ cb,
              float* __restrict__ scores) {
  __shared__ __align__(16) float lds_x2[32 + L_];     // [32+j] = x[j]
  __shared__ __align__(16) float lds_ck[NPAT_ * HIST_];
  __shared__ __align__(16) float lds_cb[NPAT_];

  const int tid = threadIdx.x;
  const int b   = blockIdx.x;

  if (tid < 32) lds_x2[tid] = 0.f;                    // left pad zeros

  // Async staging: x row (2048 x b128), ck (512 x b128), cb (16 x b128).
  const float4* xs = (const float4*)(x + (size_t)b * L_);
  float4* xd = ((float4*)lds_x2) + 8;                 // lds_x2[32]
#pragma unroll 2
  for (int i = tid; i < L_ / 4; i += 256) async_copy_b128(xd + i, xs + i);
  const float4* cks = (const float4*)ck;
  float4* ckd = (float4*)lds_ck;
  for (int i = tid; i < (NPAT_ * HIST_) / 4; i += 256) async_copy_b128(ckd + i, cks + i);
  if (tid < NPAT_ / 4) async_copy_b128(((float4*)lds_cb) + tid, ((const float4*)cb) + tid);
  wait_async0();
  __syncthreads();

  const int lane = tid & 31;
  const int wave = tid >> 5;
  const int ln   = lane & 15;          // M (for A) / N (for B) within tile
  const int kl   = (lane >> 4) << 1;   // K offset: 0 for lanes 0-15, 2 for 16-31

  // B fragments: depend only on (nt, s); hoisted.
  v2f Bf[4][8];
#pragma unroll
  for (int nt = 0; nt < 4; ++nt) {
#pragma unroll
    for (int s = 0; s < 8; ++s) {
      const int p = nt * 16 + ln;
      Bf[nt][s].x = lds_ck[p * HIST_ + s * 4 + kl];
      Bf[nt][s].y = lds_ck[p * HIST_ + s * 4 + kl + 1];
    }
  }

  float* sc = scores + (size_t)b * T_ * NPAT_;

  for (int mt = 0; mt < 8; ++mt) {
    const int t0   = wave * 128 + mt * 16;
    const int trow = t0 + ln;
    v2f Af[8];
#pragma unroll
    for (int s = 0; s < 8; ++s) {
      Af[s].x = lds_x2[trow * WIN_ + s * 4 + kl + 1];
      Af[s].y = lds_x2[trow * WIN_ + s * 4 + kl + 2];
    }
#pragma unroll
    for (int nt = 0; nt < 4; ++nt) {
      const float ci = lds_cb[nt * 16 + ln];  // bias broadcast down column N
      v8f acc = { ci, ci, ci, ci, ci, ci, ci, ci };
#pragma unroll
      for (int s = 0; s < 8; ++s) {
        acc = __builtin_amdgcn_wmma_f32_16x16x4_f32(
            /*neg_a=*/false, Af[s], /*neg_b=*/false, Bf[nt][s],
            /*c_mod=*/(short)0, acc, /*reuse_a=*/false, /*reuse_b=*/false);
      }
      const int pcol = nt * 16 + ln;
      const int tb   = t0 + ((lane >> 4) << 3);
#pragma unroll
      for (int i = 0; i < 8; ++i) {
        float v = acc[i];
        v = fminf(fmaxf(v, 0.f), 6.f);              // relu6 clip
        sc[(size_t)(tb + i) * NPAT_ + pcol] = v;
      }
    }
  }
}

// ---------------------------------------------------------------------------
// Kernel 3: sequential balanced-routing scan + one-hot decode.
// One workgroup per batch (8 waves):
//   wave 0 : serial scan, 2 patterns per lane, LDS-resident score chunks
//   wave 1 : TDM double-buffered prefetch of the next 16 KB score chunk
//            (tensor_load_to_lds + s_wait_tensorcnt), overlapped with scan
//   waves 2-7 : one-shot L2 warm of this batch's score region (prefetch),
//               then idle until the parallel decode phase
// argmax key: sortable-float bits of (score-avg), low 6 bits = (63-p)
// so ties resolve to the lowest pattern index (matches jnp.argmax).
// ---------------------------------------------------------------------------
__global__ void __launch_bounds__(256)
scan_decode_kernel(const float* __restrict__ scores,
                   const float* __restrict__ avg_init,
                   const float* __restrict__ shapes_w,
                   const float* __restrict__ x,
                   float* __restrict__ out) {
  __shared__ __align__(16) float sbuf[2][64 * NPAT_];  // 2 x 16 KB
  __shared__ int   sidx[T_];
  __shared__ float sshape[WIN_ * NPAT_];

  const int tid  = threadIdx.x;
  const int b    = blockIdx.x;
  const int lane = tid & 31;
  const int wave = tid >> 5;

  for (int i = tid; i < WIN_ * NPAT_; i += 256) sshape[i] = shapes_w[i];

  const float* src = scores + (size_t)b * T_ * NPAT_;

  // Wave 1: TDM-load chunk 0. Waves 2-7: warm L2 for the whole batch region.
  if (wave == 1) {
    tdm_load_chunk16k(&sbuf[0][0], src);
    wait_tensor0();
  } else if (wave >= 2) {
    const char* pb = (const char*)src;
    for (size_t off = (size_t)(tid - 64) * 128; off < (size_t)T_ * NPAT_ * 4;
         off += (size_t)192 * 128)
      __builtin_prefetch(pb + off, 0, 0);             // global_prefetch_b8
  }
  __syncthreads();

  // avg0 = avg_init - mean(avg_init): lane owns patterns p=lane and p=lane+32.
  float avgA = 0.f, avgB = 0.f;
  if (wave == 0) {
    const float a0 = avg_init[b * NPAT_ + lane];
    const float a1 = avg_init[b * NPAT_ + 32 + lane];
    float tot = a0 + a1;
#pragma unroll
    for (int m = 16; m > 0; m >>= 1) tot += __shfl_xor(tot, m, 32);
    const float mean = tot * (1.0f / 64.0f);
    avgA = a0 - mean;
    avgB = a1 - mean;
  }

  const int NCH = T_ / 64;  // 16 chunks of 64 steps
  for (int c = 0; c < NCH; ++c) {
    const int cur = c & 1;
    if (wave == 1) {
      if (c + 1 < NCH) {
        tdm_load_chunk16k(&sbuf[cur ^ 1][0], src + (size_t)(c + 1) * 64 * NPAT_);
        wait_tensor0();   // DMA completes while wave 0 scans the current chunk
      }
    } else if (wave == 0) {
      const float* buf = sbuf[cur];
      for (int j = 0; j < 64; ++j) {
        const float s0 = buf[j * NPAT_ + lane];
        const float s1 = buf[j * NPAT_ + 32 + lane];
        const float d0 = s0 - avgA;
        const float d1 = s1 - avgB;
        unsigned u0 = __float_as_uint(d0);
        u0 = (u0 & 0x80000000u) ? ~u0 : (u0 | 0x80000000u);
        unsigned u1 = __float_as_uint(d1);
        u1 = (u1 & 0x80000000u) ? ~u1 : (u1 | 0x80000000u);
        unsigned k0 = (u0 & ~63u) | (unsigned)(63 - lane);
        unsigned k1 = (u1 & ~63u) | (unsigned)(31 - lane);
        unsigned k  = (k0 > k1) ? k0 : k1;
#pragma unroll
        for (int m = 16; m > 0; m >>= 1) {
          const unsigned o = (unsigned)__shfl_xor((int)k, m, 32);
          k = (o > k) ? o : k;
        }
        const int p = 63 - (int)(k & 63u);
        avgA += ((p == lane)      ? 1.0f : 0.0f) - 0.015625f;  // 1/64
        avgB += ((p == lane + 32) ? 1.0f : 0.0f) - 0.015625f;
        if (lane == 0) sidx[c * 64 + j] = p;
      }
    }
    __syncthreads();
  }

  // Decode: out[b, 8t+w] = relu(shapes_w[w, idx[t]] - x[b, 8t+w])
  const float* xb = x   + (size_t)b * L_;
  float*       ob = out + (size_t)b * L_;
  for (int t = tid; t < T_; t += 256) {
    const int p = sidx[t];
    float sv[8];
#pragma unroll
    for (int w = 0; w < 8; ++w) sv[w] = sshape[w * NPAT_ + p];
    const float4 xa = ((const float4*)xb)[t * 2];
    const float4 xc = ((const float4*)xb)[t * 2 + 1];
    float4 ra, rc;
    ra.x = fmaxf(sv[0] - xa.x, 0.f);
    ra.y = fmaxf(sv[1] - xa.y, 0.f);
    ra.z = fmaxf(sv[2] - xa.z, 0.f);
    ra.w = fmaxf(sv[3] - xa.w, 0.f);
    rc.x = fmaxf(sv[4] - xc.x, 0.f);
    rc.y = fmaxf(sv[5] - xc.y, 0.f);
    rc.z = fmaxf(sv[6] - xc.z, 0.f);
    rc.w = fmaxf(sv[7] - xc.w, 0.f);
    ((float4*)ob)[t * 2]     = ra;
    ((float4*)ob)[t * 2 + 1] = rc;
  }
}

// ---------------------------------------------------------------------------
// Host launcher. Workspace layout (floats):
//   [0, 256*1024*64)            scores   (64 MB)
//   [+0, +2048)                 ck
//   [+2048, +2112)              cb
// ---------------------------------------------------------------------------
extern "C" void kernel_launch(void* const* d_in, const int* in_sizes, int n_in,
                              void* d_out, int out_size, void* d_ws, size_t ws_size,
                              hipStream_t stream) {
  (void)in_sizes; (void)n_in; (void)out_size; (void)ws_size;

  const float* x      = (const float*)d_in[0];
  const float* avgi   = (const float*)d_in[1];
  const float* conv_w = (const float*)d_in[2];
  const float* conv_b = (const float*)d_in[3];
  const float* keys_w = (const float*)d_in[4];
  const float* shapes = (const float*)d_in[5];
  float* out = (float*)d_out;

  float* scores = (float*)d_ws;
  float* ck = scores + (size_t)B_ * T_ * NPAT_;
  float* cb = ck + NPAT_ * HIST_;

  prep_kernel<<<1, 256, 0, stream>>>(keys_w, conv_w, conv_b, ck, cb);
  scores_kernel<<<B_, 256, 0, stream>>>(x, ck, cb, scores);
  scan_decode_kernel<<<B_, 256, 0, stream>>>(scores, avgi, shapes, x, out);
}